// Loss_orthogonal_48395691491826
// MI455X (gfx1250) — compile-verified
//
#include <hip/hip_runtime.h>
#include <hip/hip_bf16.h>
#include <stddef.h>

// ort = mean(x1 @ x2^T) = dot(colsum(x1), colsum(x2)) / N^2
// Stage 1 : WMMA (ones-A) column-sum reduction -> partial[2][RB][LD]
// Stage 2a: distributed fold + per-block dot partial (4 blocks)
// Stage 2b: final 4-way sum + 2^-26 scale

#define LN 8192      // N rows
#define LD 1024      // D cols
#define RB 32        // row blocks per array (each wave covers LN/RB = 256 rows)
#define STRIPS 64    // column strips of 16 (64*16 = 1024)

typedef __attribute__((ext_vector_type(2))) float v2f;
typedef __attribute__((ext_vector_type(8))) float v8f;

// Each wave: one 16-column strip x one 256-row block of one input array.
// B tile = 4 rows x 16 cols; A = ones(16x4) so D[m][n] = sum_k B[k][n] + C.
// Lanes {L, L+16} x {b.x, b.y} hold 4 distinct rows of the SAME true column,
// so the WMMA k-sum is exactly that column's 4-row sum independent of the
// hardware k/n-to-(VGPR,lane) assignment.
__global__ void __launch_bounds__(256)
colsum_wmma_kernel(const float* __restrict__ x1,
                   const float* __restrict__ x2,
                   float* __restrict__ partial /* [2][RB][LD] */)
{
    const int arr = blockIdx.y;
    const float* __restrict__ x = (arr == 0) ? x1 : x2;

    const int wave  = (blockIdx.x * blockDim.x + threadIdx.x) >> 5;
    const int lane  = threadIdx.x & 31;
    const int strip = wave & (STRIPS - 1);   // 0..63
    const int rb    = wave >> 6;             // 0..RB-1
    const int col   = strip * 16 + (lane & 15);
    const int lhalf = lane >> 4;             // 0 or 1

    const int rowsPerWave = LN / RB;         // 256
    const int r0 = rb * rowsPerWave;

    // A = ones(16x4): every lane's 2 A-VGPRs hold 1.0f.
    v2f a; a.x = 1.0f; a.y = 1.0f;
    v8f acc0 = {};
    v8f acc1 = {};

    const float* p = x + (size_t)(r0 + 2 * lhalf) * LD + col;

#pragma unroll 4
    for (int it = 0; it < rowsPerWave / 8; ++it) {
        v2f b0, b1;
        b0.x = p[0];
        b0.y = p[LD];
        b1.x = p[4 * LD];
        b1.y = p[5 * LD];
        // (neg_a, A, neg_b, B, c_mod, C, reuse_a, reuse_b)
        acc0 = __builtin_amdgcn_wmma_f32_16x16x4_f32(
            false, a, false, b0, (short)0, acc0, false, false);
        acc1 = __builtin_amdgcn_wmma_f32_16x16x4_f32(
            false, a, false, b1, (short)0, acc1, false, false);
        p += 8 * LD;
    }

    // Every accumulator row m holds the identical column sum for n = lane&15;
    // lanes L and L+16 carry the same column -> only lanes 0..15 write.
    float csum = acc0[0] + acc1[0];
    if (lane < 16) {
        partial[((size_t)arr * RB + rb) * LD + col] = csum;
    }
}

// Stage 2a: 4 blocks x 256 threads. Thread owns one column: fold RB partials
// for both arrays, multiply, then LDS tree -> one scalar per block.
__global__ void __launch_bounds__(256)
dot_partial_kernel(const float* __restrict__ partial,
                   float* __restrict__ blocksum /* [4] */)
{
    __shared__ float sdata[256];
    const int col = blockIdx.x * 256 + threadIdx.x;

    float s1 = 0.0f, s2 = 0.0f;
#pragma unroll
    for (int rb = 0; rb < RB; ++rb) {
        s1 += partial[(size_t)rb * LD + col];
        s2 += partial[((size_t)RB + rb) * LD + col];
    }
    sdata[threadIdx.x] = s1 * s2;
    __syncthreads();

#pragma unroll
    for (int s = 128; s > 0; s >>= 1) {
        if (threadIdx.x < s) sdata[threadIdx.x] += sdata[threadIdx.x + s];
        __syncthreads();
    }
    if (threadIdx.x == 0) blocksum[blockIdx.x] = sdata[0];
}

// Stage 2b: tiny final combine; 1/N^2 = 2^-26 exact.
__global__ void dot_final_kernel(const float* __restrict__ blocksum,
                                 float* __restrict__ out)
{
    if (threadIdx.x == 0) {
        float s = blocksum[0] + blocksum[1] + blocksum[2] + blocksum[3];
        out[0] = s * (1.0f / (8192.0f * 8192.0f));
    }
}

extern "C" void kernel_launch(void* const* d_in, const int* in_sizes, int n_in,
                              void* d_out, int out_size, void* d_ws, size_t ws_size,
                              hipStream_t stream) {
    const float* x1 = (const float*)d_in[0];
    const float* x2 = (const float*)d_in[1];
    float* out      = (float*)d_out;
    float* partial  = (float*)d_ws;               // 2*RB*LD floats = 256 KB
    float* blocksum = partial + 2 * RB * LD;      // +4 floats

    // Stage 1: 2048 waves per array, 8 waves per block -> 256 blocks, y = array.
    dim3 grid1((STRIPS * RB) / 8, 2, 1);
    colsum_wmma_kernel<<<grid1, 256, 0, stream>>>(x1, x2, partial);

    // Stage 2: distributed fold + tiny final combine.
    dot_partial_kernel<<<4, 256, 0, stream>>>(partial, blocksum);
    dot_final_kernel<<<1, 32, 0, stream>>>(blocksum, out);
}